// QuantizedZFPrecoder_36627481101250
// MI455X (gfx1250) — compile-verified
//
#include <hip/hip_runtime.h>
#include <math.h>

// Problem constants (from reference)
#define B_  8
#define NT_ 1
#define U_  4
#define A_  4
#define RA_ 16
#define M_  16
#define T_  14
#define F_  512
#define R_  2
#define S_  8              // U_*R_
#define NPTS (B_*T_*F_)    // 57344

// LDS layout for fused kernel: per-point stride 137 words (odd -> conflict-free
// per-thread phase-2 reads), imag plane at word offset 72 (-> phase-1 store
// lane-groups hit disjoint banks).
#define HH_STRIDE 137
#define HH_IM     72

typedef __attribute__((ext_vector_type(2))) float v2f;
typedef __attribute__((ext_vector_type(8))) float v8f;

__host__ __device__ static constexpr int trilo(int i, int j) { return i*(i+1)/2 + j; }

// ---------------------------------------------------------------------------
// Kernel 1: per (b,u,t,f) thread — 4x4 Hermitian eig (complex Jacobi) of
// h_ue h_ue^H, top-2 left singular vectors, write 2x16 h_eff rows.
// ws_heff layout per point p=(b*T+t)*F+f: 256 floats = re[8][16] then im[8][16]
// ---------------------------------------------------------------------------
__global__ __launch_bounds__(256) void zf_eig_heff(
    const float* __restrict__ hre, const float* __restrict__ him,
    const int*  __restrict__ sched, float* __restrict__ ws_heff)
{
  const int tid = blockIdx.x*blockDim.x + threadIdx.x;
  const int f = tid % F_;
  const int t = (tid / F_) % T_;
  const int u = (tid / (F_*T_)) % U_;
  const int b = tid / (F_*T_*U_);
  const int tf = t*F_ + f;

  int hoff[A_];
  #pragma unroll
  for (int a = 0; a < A_; ++a) {
    const int ra = sched[u*A_ + a];
    hoff[a] = ((b*RA_ + ra)*M_) * (T_*F_) + tf;   // + m*(T_*F_)
  }

  float wr[4][4], wi[4][4], vr[4][4], vi[4][4];
  #pragma unroll
  for (int i = 0; i < 4; ++i)
    #pragma unroll
    for (int j = 0; j < 4; ++j) {
      wr[i][j] = 0.0f; wi[i][j] = 0.0f;
      vr[i][j] = (i == j) ? 1.0f : 0.0f; vi[i][j] = 0.0f;
    }

  // Gram: W = h_ue h_ue^H  (4x4 Hermitian), streamed over m
  for (int m = 0; m < M_; ++m) {
    float cr[4], ci[4];
    #pragma unroll
    for (int a = 0; a < 4; ++a) {
      const int o = hoff[a] + m*(T_*F_);
      cr[a] = hre[o]; ci[a] = him[o];
    }
    if (m + 1 < M_) {
      __builtin_prefetch(hre + hoff[0] + (m+1)*(T_*F_), 0, 1);
      __builtin_prefetch(him + hoff[0] + (m+1)*(T_*F_), 0, 1);
    }
    #pragma unroll
    for (int p = 0; p < 4; ++p)
      #pragma unroll
      for (int q = p; q < 4; ++q) {
        wr[p][q] += cr[p]*cr[q] + ci[p]*ci[q];
        wi[p][q] += ci[p]*cr[q] - cr[p]*ci[q];
      }
  }
  #pragma unroll
  for (int p = 0; p < 4; ++p)
    #pragma unroll
    for (int q = p+1; q < 4; ++q) { wr[q][p] = wr[p][q]; wi[q][p] = -wi[p][q]; }

  // Complex Jacobi sweeps (sweep loop kept rolled to bound code size)
  for (int sweep = 0; sweep < 6; ++sweep) {
    #pragma unroll
    for (int p = 0; p < 3; ++p)
      #pragma unroll
      for (int q = p+1; q < 4; ++q) {
        const float apr = wr[p][q], api = wi[p][q];
        const float m2 = apr*apr + api*api;
        if (m2 > 1e-24f) {
          const float mag = sqrtf(m2);
          const float er = apr/mag, ei = api/mag;
          const float tau = (wr[q][q] - wr[p][p]) / (2.0f*mag);
          const float ts = (tau >= 0.0f) ? 1.0f : -1.0f;
          const float tt = ts / (fabsf(tau) + sqrtf(1.0f + tau*tau));
          const float c  = 1.0f / sqrtf(1.0f + tt*tt);
          const float s  = tt*c;
          // W <- W * J (columns p,q)
          #pragma unroll
          for (int j = 0; j < 4; ++j) {
            const float pr = wr[j][p], pi = wi[j][p], qr = wr[j][q], qi = wi[j][q];
            wr[j][p] = c*pr + s*(er*qr + ei*qi);
            wi[j][p] = c*pi + s*(er*qi - ei*qr);
            wr[j][q] = c*qr - s*(er*pr - ei*pi);
            wi[j][q] = c*qi - s*(er*pi + ei*pr);
          }
          // W <- J^H * W (rows p,q)
          #pragma unroll
          for (int j = 0; j < 4; ++j) {
            const float pr = wr[p][j], pi = wi[p][j], qr = wr[q][j], qi = wi[q][j];
            wr[p][j] = c*pr + s*(er*qr - ei*qi);
            wi[p][j] = c*pi + s*(er*qi + ei*qr);
            wr[q][j] = c*qr - s*(er*pr + ei*pi);
            wi[q][j] = c*qi - s*(er*pi - ei*pr);
          }
          // V <- V * J
          #pragma unroll
          for (int j = 0; j < 4; ++j) {
            const float pr = vr[j][p], pi = vi[j][p], qr = vr[j][q], qi = vi[j][q];
            vr[j][p] = c*pr + s*(er*qr + ei*qi);
            vi[j][p] = c*pi + s*(er*qi - ei*qr);
            vr[j][q] = c*qr - s*(er*pr - ei*pi);
            vi[j][q] = c*qi - s*(er*pi + ei*pr);
          }
        }
      }
  }

  // pick top-2 eigenvalues (descending, matches SVD ordering)
  const float d0 = wr[0][0], d1 = wr[1][1], d2 = wr[2][2], d3 = wr[3][3];
  int i0 = 0; float best = d0;
  if (d1 > best) { best = d1; i0 = 1; }
  if (d2 > best) { best = d2; i0 = 2; }
  if (d3 > best) { best = d3; i0 = 3; }
  int i1 = 0; float best2 = -3.0e38f;
  #pragma unroll
  for (int k = 0; k < 4; ++k) {
    const float dk = (k==0)?d0:(k==1)?d1:(k==2)?d2:d3;
    if (k != i0 && dk > best2) { best2 = dk; i1 = k; }
  }

  float u0r[4], u0i[4], u1r[4], u1i[4];
  #pragma unroll
  for (int a = 0; a < 4; ++a) {
    u0r[a] = (i0==0)?vr[a][0]:(i0==1)?vr[a][1]:(i0==2)?vr[a][2]:vr[a][3];
    u0i[a] = (i0==0)?vi[a][0]:(i0==1)?vi[a][1]:(i0==2)?vi[a][2]:vi[a][3];
    u1r[a] = (i1==0)?vr[a][0]:(i1==1)?vr[a][1]:(i1==2)?vr[a][2]:vr[a][3];
    u1i[a] = (i1==0)?vi[a][0]:(i1==1)?vi[a][1]:(i1==2)?vi[a][2]:vi[a][3];
  }

  // h_eff[r][m] = sum_a conj(Ur[a][r]) * h_ue[a][m]  (stream m, L2-hot reload)
  const int pnt = (b*T_ + t)*F_ + f;
  float* outp = ws_heff + (size_t)pnt*256;
  const int s0 = u*2, s1 = u*2 + 1;
  for (int m = 0; m < M_; ++m) {
    float cr[4], ci[4];
    #pragma unroll
    for (int a = 0; a < 4; ++a) {
      const int o = hoff[a] + m*(T_*F_);
      cr[a] = hre[o]; ci[a] = him[o];
    }
    float e0r = 0.f, e0i = 0.f, e1r = 0.f, e1i = 0.f;
    #pragma unroll
    for (int a = 0; a < 4; ++a) {
      e0r += u0r[a]*cr[a] + u0i[a]*ci[a];
      e0i += u0r[a]*ci[a] - u0i[a]*cr[a];
      e1r += u1r[a]*cr[a] + u1i[a]*ci[a];
      e1i += u1r[a]*ci[a] - u1i[a]*cr[a];
    }
    outp[s0*16 + m] = e0r; outp[128 + s0*16 + m] = e0i;
    outp[s1*16 + m] = e1r; outp[128 + s1*16 + m] = e1i;
  }
}

// ---------------------------------------------------------------------------
// Kernel 2 (fused): 256-thread block handles 256 points.
// Phase 1: each of the 8 waves computes 32 Grams hhH = h_eff h_eff^H via the
//   real embedding Hhat (16x32): Hhat*Hhat^T = [[Re,-Im],[Im,Re]] using
//   8 chained V_WMMA_F32_16X16X4_F32 (A register image doubles as B for A*A^T),
//   scattering Re/Im 8x8 into LDS.
// Phase 2: after the workgroup barrier, each thread runs complex Cholesky,
//   L^-1, W = L^-H L^-1 for one point (hhH from LDS), derives column norms
//   from diag(W), and streams out normalized g and the precoded x.
// ---------------------------------------------------------------------------
__global__ __launch_bounds__(256) void zf_gram_solve_apply(
    const float* __restrict__ ws_heff,
    const float* __restrict__ xre, const float* __restrict__ xim,
    float* __restrict__ out_x, float* __restrict__ out_g)
{
  extern __shared__ float hh[];             // 256 * HH_STRIDE floats (~140 KB)

  const int wv   = threadIdx.x >> 5;
  const int lane = threadIdx.x & 31;
  const int pbase = blockIdx.x * 256;

  // -------- Phase 1: WMMA Grams into LDS --------
  {
    const int Mrow = lane & 15;             // A layout: lane holds row M
    const int half = lane >> 4;             // upper half-lanes hold K+2,K+3
    const int srow = Mrow & 7;
    const bool bot = (Mrow >= 8);
    const int col = lane & 15;

    for (int i = 0; i < 32; ++i) {
      const int pt = pbase + wv*32 + i;
      const float* hp = ws_heff + (size_t)pt*256;
      v8f acc = {};
      #pragma unroll
      for (int c = 0; c < 8; ++c) {
        const int K0 = 4*c + 2*half;        // even -> (K0,K0+1) in one quadrant
        const bool right = (K0 >= 16);
        const int k = K0 & 15;
        // quadrant select: TL->Hr, TR->-Hi, BL->Hi, BR->Hr
        const float* plane = (bot != right) ? (hp + 128) : hp;
        v2f ab = *(const v2f*)(plane + srow*16 + k);
        const float sgn = (!bot && right) ? -1.0f : 1.0f;
        ab.x *= sgn; ab.y *= sgn;
        acc = __builtin_amdgcn_wmma_f32_16x16x4_f32(
            false, ab, false, ab, (short)0, acc, false, false);
      }
      // D: vgpr r, lane l -> row (r + 8*half), col (l&15).
      // rows 0..7 = Re(hhH) -> LDS plane 0; rows 8..15 = Im -> plane HH_IM.
      if (col < 8) {
        float* dst = &hh[(wv*32 + i)*HH_STRIDE + half*HH_IM + col];
        #pragma unroll
        for (int r = 0; r < 8; ++r) dst[r*8] = acc[r];
      }
    }
  }

  __syncthreads();

  // -------- Phase 2: per-thread solve + apply --------
  const int p = pbase + threadIdx.x;
  const int f = p % F_;
  const int t = (p / F_) % T_;
  const int b = p / (F_*T_);
  const float* hhp = &hh[threadIdx.x * HH_STRIDE];

  // packed lower triangle of Hermitian hhH (36 complex) from LDS
  float lr[36], li[36];
  #pragma unroll
  for (int i = 0; i < 8; ++i)
    #pragma unroll
    for (int j = 0; j <= i; ++j) {
      lr[trilo(i,j)] = hhp[i*8 + j];
      li[trilo(i,j)] = hhp[HH_IM + i*8 + j];
    }

  // complex Cholesky: hhH = L L^H
  float dinv[8];
  #pragma unroll
  for (int j = 0; j < 8; ++j) {
    float d = lr[trilo(j,j)];
    #pragma unroll
    for (int k = 0; k < j; ++k)
      d -= lr[trilo(j,k)]*lr[trilo(j,k)] + li[trilo(j,k)]*li[trilo(j,k)];
    d = sqrtf(fmaxf(d, 1e-30f));
    lr[trilo(j,j)] = d;
    const float idj = 1.0f / d;
    dinv[j] = idj;
    #pragma unroll
    for (int i = j+1; i < 8; ++i) {
      float sr = lr[trilo(i,j)], si = li[trilo(i,j)];
      #pragma unroll
      for (int k = 0; k < j; ++k) {
        const float ar = lr[trilo(i,k)], ai = li[trilo(i,k)];
        const float br = lr[trilo(j,k)], bi = li[trilo(j,k)];
        sr -= ar*br + ai*bi;            // -= L[i][k]*conj(L[j][k])
        si -= ai*br - ar*bi;
      }
      lr[trilo(i,j)] = sr*idj; li[trilo(i,j)] = si*idj;
    }
  }

  // invert L in place (off-diag); diag(L^-1) kept in dinv[]
  #pragma unroll
  for (int j = 0; j < 8; ++j)
    #pragma unroll
    for (int i = j+1; i < 8; ++i) {
      float tr = lr[trilo(i,j)]*dinv[j];
      float ti = li[trilo(i,j)]*dinv[j];
      #pragma unroll
      for (int k = j+1; k < i; ++k) {
        const float ar = lr[trilo(i,k)], ai = li[trilo(i,k)];  // original L
        const float br = lr[trilo(k,j)], bi = li[trilo(k,j)];  // Linv col j
        tr += ar*br - ai*bi;
        ti += ar*bi + ai*br;
      }
      lr[trilo(i,j)] = -tr*dinv[i];
      li[trilo(i,j)] = -ti*dinv[i];
    }

  // W = Linv^H Linv (Hermitian, packed lower: s >= k)
  float wr2[36], wi2[36];
  #pragma unroll
  for (int s = 0; s < 8; ++s)
    #pragma unroll
    for (int k = 0; k <= s; ++k) {
      float ar, ai;
      if (s == k) { ar = dinv[s]*dinv[s]; ai = 0.0f; }
      else        { ar = dinv[s]*lr[trilo(s,k)]; ai = dinv[s]*li[trilo(s,k)]; }
      #pragma unroll
      for (int j = s+1; j < 8; ++j) {
        const float br = lr[trilo(j,s)], bi = li[trilo(j,s)];  // Linv[j][s]
        const float cr = lr[trilo(j,k)], ci = li[trilo(j,k)];  // Linv[j][k]
        ar += br*cr + bi*ci;            // conj(Linv[j][s]) * Linv[j][k]
        ai += br*ci - bi*cr;
      }
      wr2[trilo(s,k)] = ar; wi2[trilo(s,k)] = ai;
    }

  // column norms of g are sqrt(diag(W)) analytically
  float inrm[8];
  #pragma unroll
  for (int k = 0; k < 8; ++k) inrm[k] = rsqrtf(fmaxf(wr2[trilo(k,k)], 1e-30f));

  float xr_[8], xi_[8];
  #pragma unroll
  for (int s = 0; s < 8; ++s) {
    const int xo = ((b*S_ + s)*T_ + t)*F_ + f;
    xr_[s] = xre[xo]; xi_[s] = xim[xo];
  }

  const float* hf = ws_heff + (size_t)p*256;
  for (int m = 0; m < M_; ++m) {
    float hcr[8], hci[8];
    #pragma unroll
    for (int s = 0; s < 8; ++s) { hcr[s] = hf[s*16 + m]; hci[s] = hf[128 + s*16 + m]; }
    float yr = 0.0f, yi = 0.0f;
    #pragma unroll
    for (int k = 0; k < 8; ++k) {
      float gr = 0.0f, gi = 0.0f;
      #pragma unroll
      for (int s = 0; s < 8; ++s) {
        float wrv, wiv;
        if (s >= k) { wrv = wr2[trilo(s,k)]; wiv =  wi2[trilo(s,k)]; }
        else        { wrv = wr2[trilo(k,s)]; wiv = -wi2[trilo(k,s)]; }
        gr += hcr[s]*wrv + hci[s]*wiv;  // conj(h_eff[s][m]) * W[s][k]
        gi += hcr[s]*wiv - hci[s]*wrv;
      }
      gr *= inrm[k]; gi *= inrm[k];
      const int go = ((p*16 + m)*8 + k)*2;    // g: (b,t,f,m,k) interleaved
      out_g[go] = gr; out_g[go + 1] = gi;
      yr += gr*xr_[k] - gi*xi_[k];            // y[m] = sum_s g[m,s] x[s]
      yi += gr*xi_[k] + gi*xr_[k];
    }
    const int xo = ((b*M_ + m)*T_ + t)*F_ + f;  // x_precoded: (b,0,m,t,f)
    out_x[2*xo] = yr; out_x[2*xo + 1] = yi;
  }
}

// ---------------------------------------------------------------------------
extern "C" void kernel_launch(void* const* d_in, const int* in_sizes, int n_in,
                              void* d_out, int out_size, void* d_ws, size_t ws_size,
                              hipStream_t stream)
{
  (void)in_sizes; (void)n_in; (void)out_size; (void)ws_size;
  const float* xr    = (const float*)d_in[0];   // x_rg_real  (B,NT,S,T,F)
  const float* xi    = (const float*)d_in[1];   // x_rg_imag
  const float* hr    = (const float*)d_in[2];   // h_real     (B,RA,M,T,F)
  const float* hi    = (const float*)d_in[3];   // h_imag
  const int*   sched = (const int*)d_in[4];     // (U,A)
  // d_in[5] = ue_rank (compile-time R_=2)

  float* ws_heff = (float*)d_ws;                          // NPTS * 256 f32

  float* out_x = (float*)d_out;                           // (B,NT,M,T,F) cplx
  float* out_g = out_x + 2*(size_t)B_*NT_*M_*T_*F_;       // (B,T,F,M,S) cplx

  const size_t lds_bytes = (size_t)256 * HH_STRIDE * sizeof(float);

  zf_eig_heff<<<(B_*U_*T_*F_)/256, 256, 0, stream>>>(hr, hi, sched, ws_heff);
  zf_gram_solve_apply<<<NPTS/256, 256, lds_bytes, stream>>>(ws_heff, xr, xi,
                                                            out_x, out_g);
}